// OneVectorProjector_3968549781677
// MI455X (gfx1250) — compile-verified
//
#include <hip/hip_runtime.h>

#define OUT_FEATS 4096
#define IN_FEATS  16384
#define EPS 1e-8f

typedef __attribute__((ext_vector_type(2))) float v2f;
typedef __attribute__((ext_vector_type(4))) float f4;
typedef __attribute__((ext_vector_type(8))) float v8f;

// ws layout (floats): [0 .. IN_FEATS-1] = colsum accumulator, [IN_FEATS] = 1/(||v||^2+eps)

// Pass 0: denom reduction + zero colsum (single block; tiny).
__global__ void proj_pass0(const float* __restrict__ v, float* __restrict__ ws) {
    __shared__ float red[256];
    const int tid = threadIdx.x;
    float s = 0.f;
    for (int i = tid; i < OUT_FEATS; i += 256) { float x = v[i]; s += x * x; }
    red[tid] = s;
    __syncthreads();
    for (int off = 128; off > 0; off >>= 1) {
        if (tid < off) red[tid] += red[tid + off];
        __syncthreads();
    }
    if (tid == 0) ws[IN_FEATS] = 1.0f / (red[0] + EPS);
    for (int i = tid; i < IN_FEATS; i += 256) ws[i] = 0.0f;
}

// Pass 1: colsum[j] = sum_i v[i]*A[i,j] via v_wmma_f32_16x16x4_f32.
// A-operand (16x4, M x K): all 16 rows = v[i..i+3]  (broadcast).
//   lanes 0-15:  vgpr0=K0, vgpr1=K1 ; lanes 16-31: vgpr0=K2, vgpr1=K3.
// B-operand (4x16, K x N): B[k,n] = A[i+k, j0+n], N striped across lanes:
//   lanes 0-15:  vgpr0=(K0,N=l), vgpr1=(K1,N=l) ; lanes 16-31: K2/K3.
// => per lane: two dword loads from rows (i+half2, i+half2+1), same column.
// D[m,n] identical over m; lanes 0-15 read D[0,n] from acc vgpr0.
// Grid: x = IN_FEATS/16 column tiles, y = 8 row chunks; 4 waves/block,
// each wave reduces 128 rows (32 WMMAs), then atomically adds 16 partials.
__global__ void __launch_bounds__(128)
proj_pass1(const float* __restrict__ A, const float* __restrict__ v,
           float* __restrict__ colsum) {
    const int lane  = threadIdx.x & 31;
    const int wave  = threadIdx.x >> 5;
    const int l     = lane & 15;          // N / column within tile
    const int half2 = (lane >> 4) << 1;   // 0 for lanes 0-15, 2 for lanes 16-31
    const int j0    = blockIdx.x * 16;
    const int r0    = (blockIdx.y * 4 + wave) * 128;

    const float* __restrict__ colp = A + (size_t)j0 + (size_t)l;

    v8f acc = {};
#pragma unroll 4
    for (int it = 0; it < 32; ++it) {
        const int i = r0 + it * 4 + half2;
        v2f a, b;
        a.x = v[i];
        a.y = v[i + 1];
        b.x = colp[(size_t)i * IN_FEATS];
        b.y = colp[(size_t)(i + 1) * IN_FEATS];
        acc = __builtin_amdgcn_wmma_f32_16x16x4_f32(
            /*neg_a=*/false, a, /*neg_b=*/false, b,
            /*c_mod=*/(short)0, acc, /*reuse_a=*/false, /*reuse_b=*/false);
    }
    if (lane < 16) atomicAdd(&colsum[j0 + l], acc[0]);
}

// Pass 2: out[i,j] = A[i,j] - v[i] * colsum[j] * inv_denom.  Pure stream:
// b128 loads, non-temporal b128 stores (out is never re-read).
__global__ void __launch_bounds__(256)
proj_pass2(const float* __restrict__ A, const float* __restrict__ v,
           const float* __restrict__ ws, float* __restrict__ out) {
    const float inv_denom = ws[IN_FEATS];
    const size_t idx = (size_t)blockIdx.x * 256 + threadIdx.x; // float4 index
    const int row = (int)(idx >> 12);     // IN_FEATS/4 = 4096 float4 per row
    const int c4  = (int)(idx & 4095);

    const f4* __restrict__ A4 = (const f4*)A;
    const f4* __restrict__ C4 = (const f4*)ws;  // colsum
    f4* __restrict__ O4 = (f4*)out;

    const f4 a  = A4[idx];
    const f4 cs = C4[c4];
    const float s = v[row] * inv_denom;

    f4 o;
    o.x = a.x - s * cs.x;
    o.y = a.y - s * cs.y;
    o.z = a.z - s * cs.z;
    o.w = a.w - s * cs.w;
    __builtin_nontemporal_store(o, O4 + idx);
}

extern "C" void kernel_launch(void* const* d_in, const int* in_sizes, int n_in,
                              void* d_out, int out_size, void* d_ws, size_t ws_size,
                              hipStream_t stream) {
    const float* A = (const float*)d_in[0];   // raw_update [4096,16384] fp32
    const float* v = (const float*)d_in[1];   // v [4096] fp32
    float* out = (float*)d_out;
    float* ws  = (float*)d_ws;                // needs (IN_FEATS+1)*4 = 64 KiB + 4 B

    proj_pass0<<<1, 256, 0, stream>>>(v, ws);
    proj_pass1<<<dim3(IN_FEATS / 16, OUT_FEATS / 512), 128, 0, stream>>>(A, v, ws);
    proj_pass2<<<(OUT_FEATS * (IN_FEATS / 4)) / 256, 256, 0, stream>>>(A, v, ws, out);
}